// DEKRPoseEstimationDecodeCallback_24687472017771
// MI455X (gfx1250) — compile-verified
//
#include <hip/hip_runtime.h>
#include <hip/hip_bf16.h>
#include <math.h>

#define HM_C 18
#define NJ 17
#define H 256
#define W 256
#define HW (H*W)
#define NP 30
#define KPT_THR 0.01f
#define NMS_THR 0.05f
#define NMS_NUM_THR 8

typedef float v2f __attribute__((ext_vector_type(2)));
typedef float v8f __attribute__((ext_vector_type(8)));

__device__ __forceinline__ float sigmoidf(float x) { return 1.0f / (1.0f + expf(-x)); }

// ---------------------------------------------------------------------------
// Kernel 1: peaks of center channel. max-pool(5x5, pad -inf) on RAW logits
// (sigmoid is strictly monotone, so argmax identical), sigmoid only the peak.
// ---------------------------------------------------------------------------
__global__ void peaks_kernel(const float* __restrict__ hm, float* __restrict__ peaks) {
    __shared__ float tile[20][20];
    const int bx = blockIdx.x, by = blockIdx.y;
    const int tx = threadIdx.x, ty = threadIdx.y;
    const int t = ty * 16 + tx;
    const float* center = hm + 17 * HW;
    for (int e = t; e < 400; e += 256) {
        int ly = e / 20, lx = e % 20;
        int gy = by * 16 - 2 + ly;
        int gx = bx * 16 - 2 + lx;
        float v = -INFINITY;
        if (gy >= 0 && gy < H && gx >= 0 && gx < W) v = center[gy * W + gx];
        tile[ly][lx] = v;
    }
    __syncthreads();
    float raw = tile[ty + 2][tx + 2];
    float m = -INFINITY;
#pragma unroll
    for (int dy = 0; dy < 5; dy++)
#pragma unroll
        for (int dx = 0; dx < 5; dx++)
            m = fmaxf(m, tile[ty + dy][tx + dx]);
    int gy = by * 16 + ty, gx = bx * 16 + tx;
    peaks[gy * W + gx] = (raw == m) ? sigmoidf(raw) : 0.0f;
}

// ---------------------------------------------------------------------------
// Kernel 2: exact top-30 over 65536 values, one block of 1024 threads.
// Each value -> strictly-unique orderable u64 key (monotone float bits in the
// high word, ~index in the low word -> ties break to lowest index, same as
// jax.lax.top_k). Iteration i selects the max key strictly below the previous
// winner, so no removal writes are needed (values stay in registers).
// ---------------------------------------------------------------------------
__global__ void topk_kernel(const float* __restrict__ peaks,
                            float* __restrict__ scores, int* __restrict__ inds) {
    __shared__ unsigned long long red[1024];
    const int t = threadIdx.x;
    float v[64];
#pragma unroll
    for (int s = 0; s < 64; s++) v[s] = peaks[s * 1024 + t];

    unsigned long long prev = ~0ULL;
    for (int iter = 0; iter < NP; iter++) {
        unsigned long long best = 0;
#pragma unroll
        for (int s = 0; s < 64; s++) {
            unsigned u = __float_as_uint(v[s]);
            u = (u & 0x80000000u) ? ~u : (u | 0x80000000u);
            unsigned idx = (unsigned)(s * 1024 + t);
            unsigned long long key =
                ((unsigned long long)u << 32) | (unsigned long long)(0xFFFFFFFFu - idx);
            if (key < prev && key > best) best = key;
        }
        red[t] = best;
        __syncthreads();
        for (int s = 512; s > 0; s >>= 1) {
            if (t < s) {
                unsigned long long o = red[t + s];
                if (o > red[t]) red[t] = o;
            }
            __syncthreads();
        }
        unsigned long long win = red[0];
        if (t == 0) {
            unsigned widx = 0xFFFFFFFFu - (unsigned)(win & 0xFFFFFFFFu);
            unsigned uv = (unsigned)(win >> 32);
            float wval = (uv & 0x80000000u) ? __uint_as_float(uv & 0x7FFFFFFFu)
                                            : __uint_as_float(~uv);
            scores[iter] = wval;
            inds[iter] = (int)widx;
        }
        prev = win;
        __syncthreads();
    }
}

// ---------------------------------------------------------------------------
// Kernel 3: decode + NMS + outputs. One block, 128 threads (4 wave32).
// The 18x1024x1024 upsampled heatmap is never materialized: bilinear taps are
// evaluated on the fly (y index k of hm_sum == sample at min(k,1020)/4).
// Per-joint 30x30 distance matrices computed via V_WMMA_F32_16X16X4_F32 Gram
// products on 512-centered coordinates: 4 waves x 1 tile each x 17 joints.
// Rows 30..31 of sP are kept zero so the K=2/3 lane half and the matrix
// padding come for free -> no divergence anywhere around the WMMA.
// ---------------------------------------------------------------------------
__global__ void decode_kernel(const float* __restrict__ hm, const float* __restrict__ off,
                              const float* __restrict__ scores, const int* __restrict__ inds,
                              float* __restrict__ out) {
    __shared__ float sP[32][36];              // centered (x,y) pairs, zero padded
    __shared__ float sSq[NJ][32];             // per-joint squared norms, zero padded
    __shared__ float sPX[NP][NJ], sPY[NP][NJ], sHeat[NP][NJ];
    __shared__ float sScore[NP], sHS[NP], sHSM[NP];
    __shared__ float sThreP[32];              // padded: slots 30,31 = 0
    __shared__ int   sValidP[32];             // padded: slots 30,31 = 0
    __shared__ int   sInd[NP];
    __shared__ int   sNms[32][32];
    __shared__ int   sKeep[NP];

    const int t = threadIdx.x;  // 128 threads

    for (int e = t; e < 32 * 36; e += 128) ((float*)sP)[e] = 0.0f;
    for (int e = t; e < NJ * 32; e += 128) ((float*)sSq)[e] = 0.0f;
    if (t < 32) { sValidP[t] = 0; sThreP[t] = 0.0f; }
    if (t < NP) {
        float sc = scores[t];
        sScore[t] = sc;
        sInd[t] = inds[t];
        sValidP[t] = (sc > KPT_THR) ? 1 : 0;
    }
    __syncthreads();

    // ---- pose + bilinear heat value per (person, joint) ----
    for (int p = t; p < NP * NJ; p += 128) {
        int m = p / NJ, j = p % NJ;
        int ind = sInd[m];
        int py = ind >> 8, px = ind & 255;
        float ox = off[(2 * j) * HW + ind];
        float oy = off[(2 * j + 1) * HW + ind];
        float posex = 4.0f * ((float)px - ox);
        float posey = 4.0f * ((float)py - oy);
        sPX[m][j] = posex;
        sPY[m][j] = posey;
        float cx = posex - 512.0f, cy = posey - 512.0f;
        sP[m][2 * j]     = cx;
        sP[m][2 * j + 1] = cy;
        sSq[j][m] = cx * cx + cy * cy;

        int xl = (int)floorf(posex); xl = min(max(xl, 0), 1023);
        int yb = (int)floorf(posey); yb = min(max(yb, 0), 1023);
        int xc = min(xl, 1020), yc = min(yb, 1020);   // edge padding of hm_sum
        int ix0 = xc >> 2, iy0 = yc >> 2;
        float wx = (float)(xc & 3) * 0.25f, wy = (float)(yc & 3) * 0.25f;
        int ix1 = min(ix0 + 1, 255), iy1 = min(iy0 + 1, 255);
        const float* hj = hm + j * HW;
        float h00 = sigmoidf(hj[iy0 * W + ix0]);
        float h01 = sigmoidf(hj[iy0 * W + ix1]);
        float h10 = sigmoidf(hj[iy1 * W + ix0]);
        float h11 = sigmoidf(hj[iy1 * W + ix1]);
        float cL = h00 + wy * (h10 - h00);   // interp H first (matches ref order)
        float cR = h01 + wy * (h11 - h01);
        sHeat[m][j] = cL + wx * (cR - cL);
    }
    __syncthreads();

    // ---- per-person heat score, area, nms threshold ----
    if (t < NP) {
        float s = 0.0f;
        float mnx = INFINITY, mxx = -INFINITY, mny = INFINITY, mxy = -INFINITY;
        for (int j = 0; j < NJ; j++) {
            s += sHeat[t][j];
            float x = sPX[t][j], y = sPY[t][j];
            mnx = fminf(mnx, x); mxx = fmaxf(mxx, x);
            mny = fminf(mny, y); mxy = fmaxf(mxy, y);
        }
        float hs = s * (1.0f / 17.0f);
        sHS[t] = hs;
        sHSM[t] = sValidP[t] ? hs : -INFINITY;
        float area = (mxx - mnx) * (mxx - mnx) + (mxy - mny) * (mxy - mny);
        sThreP[t] = NMS_THR * sqrtf(area);
    }
    __syncthreads();

    // ---- per-joint 30x30 distance counts via WMMA Gram products ----
    {
        const int wave = t >> 5, lane = t & 31;
        const int mi = wave >> 1, ni = wave & 1;     // output tile (16x16)
        const int hi = (lane >> 4) & 1;              // lane half: K0/K1 vs K2/K3
        const int lrow = lane & 15;
        const int jj = ni * 16 + lrow;               // this lane's output column

        // lanes 16..31 must supply the zero K=2/3 slots: read the all-zero
        // pad row 31 instead of branching -> EXEC stays all-ones throughout.
        const int arow = hi ? 31 : (mi * 16 + lrow);
        const int brow = hi ? 31 : (ni * 16 + lrow);

        int cnt[8];
#pragma unroll
        for (int r = 0; r < 8; r++) cnt[r] = 0;

        for (int j = 0; j < NJ; j++) {
            v2f a, b;
            a.x = sP[arow][2 * j];
            a.y = sP[arow][2 * j + 1];
            b.x = sP[brow][2 * j];
            b.y = sP[brow][2 * j + 1];

            v8f c = {0.0f, 0.0f, 0.0f, 0.0f, 0.0f, 0.0f, 0.0f, 0.0f};
            c = __builtin_amdgcn_wmma_f32_16x16x4_f32(
                    false, a, false, b, (short)0, c, false, false);

            float sqj = sSq[j][jj];
#pragma unroll
            for (int r = 0; r < 8; r++) {
                int i = mi * 16 + r + hi * 8;        // C/D layout: vgpr r + lane half -> row
                float d2 = sSq[j][i] + sqj - 2.0f * c[r];
                float dist = sqrtf(fmaxf(d2, 0.0f));
                cnt[r] += (dist < sThreP[i]) ? 1 : 0;   // thre pad = 0 -> never true
            }
        }
#pragma unroll
        for (int r = 0; r < 8; r++) {
            int i = mi * 16 + r + hi * 8;
            // validP pad rows/cols are 0, subsuming the i<30 / jj<30 masks
            sNms[i][jj] = (cnt[r] > NMS_NUM_THR) && sValidP[i] && sValidP[jj];
        }
    }
    __syncthreads();

    // ---- greedy suppression loop (sequential, matches fori_loop body) ----
    if (t == 0) {
        int ignored[NP];
        for (int i = 0; i < NP; i++) { ignored[i] = !sValidP[i]; sKeep[i] = 0; }
        for (int i = 0; i < NP; i++) {
            int has = 0, ki = 0;
            float best = -INFINITY;
            for (int j2 = 0; j2 < NP; j2++) {
                if (sNms[i][j2]) {
                    has = 1;
                    float v = sHSM[j2];
                    if (v > best) { best = v; ki = j2; }   // first max, like argmax
                }
            }
            int cond = (!ignored[i]) && has && (!ignored[ki]);
            if (cond) {
                sKeep[ki] = 1;
                for (int j2 = 0; j2 < NP; j2++) ignored[j2] |= sNms[i][j2];
            }
        }
    }
    __syncthreads();

    // ---- outputs: poses_out (30*17*3), pose_scores (30), keep (30) ----
    for (int p = t; p < NP * NJ; p += 128) {
        int m = p / NJ, j = p % NJ;
        float k = sKeep[m] ? 1.0f : 0.0f;
        out[p * 3 + 0] = k * sPX[m][j];
        out[p * 3 + 1] = k * sPY[m][j];
        out[p * 3 + 2] = k * (sScore[m] * sHeat[m][j]);
    }
    if (t < NP) {
        float k = sKeep[t] ? 1.0f : 0.0f;
        out[NP * NJ * 3 + t] = k * (sScore[t] * sHS[t]);   // pose_scores
        out[NP * NJ * 3 + NP + t] = k;                     // keep
    }
}

extern "C" void kernel_launch(void* const* d_in, const int* in_sizes, int n_in,
                              void* d_out, int out_size, void* d_ws, size_t ws_size,
                              hipStream_t stream) {
    const float* hm  = (const float*)d_in[0];   // (1,18,256,256)
    const float* off = (const float*)d_in[1];   // (1,34,256,256)

    float* peaks  = (float*)d_ws;               // 65536 floats
    float* scores = peaks + HW;                 // 30 floats
    int*   inds   = (int*)(scores + NP);        // 30 ints

    dim3 gA(16, 16), bA(16, 16);
    peaks_kernel<<<gA, bA, 0, stream>>>(hm, peaks);
    topk_kernel<<<1, 1024, 0, stream>>>(peaks, scores, inds);
    decode_kernel<<<1, 128, 0, stream>>>(hm, off, scores, inds, (float*)d_out);
}